// ParagraphEncoder_34986803593597
// MI455X (gfx1250) — compile-verified
//
#include <hip/hip_runtime.h>
#include <math.h>

// ---------------------------------------------------------------------------
// ParagraphEncoder on MI455X (gfx1250): 2-layer bi-GRU + attention pooling.
// All matmuls via V_WMMA_F32_16X16X4_F32 (f32 in/out, matches f32 reference).
// Global accesses forced to address-space(1) so they lower to global_load/
// global_store (separate LOADcnt/STOREcnt) instead of flat (LOADcnt+DScnt).
// ---------------------------------------------------------------------------

typedef float v2f __attribute__((ext_vector_type(2)));
typedef float v8f __attribute__((ext_vector_type(8)));

#define T_SEQ 256
#define BATCH 64
#define HID   384
#define DIM   768
#define G3    1152       // 3*HID
#define HPAD  388        // padded LDS row stride for h (avoids bank conflicts)

// ---- global (AS1) memory helpers ------------------------------------------
__device__ __forceinline__ v2f ldg_v2f(const float* p) {
  return *(const __attribute__((address_space(1))) v2f*)p;
}
__device__ __forceinline__ float ldg_f(const float* p) {
  return *(const __attribute__((address_space(1))) float*)p;
}
__device__ __forceinline__ void stg_f(float* p, float v) {
  *(__attribute__((address_space(1))) float*)p = v;
}

__device__ __forceinline__ v8f wmma4(v2f a, v2f b, v8f c) {
  // D = A(16x4,f32) * B(4x16,f32) + C(16x16,f32)
  return __builtin_amdgcn_wmma_f32_16x16x4_f32(
      /*neg_a=*/false, a, /*neg_b=*/false, b,
      /*c_mod=*/(short)0, c, /*reuse_a=*/false, /*reuse_b=*/false);
}

// ---------------------------------------------------------------------------
// C[M,N] = A[M,K] @ W[N,K]^T + bias[N], optional tanh.
// Block: 256 threads = 8 waves, each wave owns a 64x64 tile (4x4 WMMA accs).
// Block tile: 256(M) x 128(N). Grid: (M/256, N/128). M,N,K multiples assumed.
// B operand: B[k][n] = W[n][k]  -> contiguous float2 along W rows.
// ---------------------------------------------------------------------------
__global__ __launch_bounds__(256) void gemm_bias_act(
    const float* __restrict__ A, const float* __restrict__ W,
    const float* __restrict__ bias, float* __restrict__ C,
    int K, int N, int act)
{
  const int lane = threadIdx.x & 31;
  const int wave = threadIdx.x >> 5;
  const int ln   = lane & 15;   // row (A) / col (B,C) within tile
  const int half = lane >> 4;   // K split for A/B, M split for C
  const int Moff = blockIdx.x * 256 + (wave >> 1) * 64;
  const int Noff = blockIdx.y * 128 + (wave & 1) * 64;

  const v8f z8 = {0,0,0,0,0,0,0,0};
  v8f acc[4][4];
#pragma unroll
  for (int mi = 0; mi < 4; ++mi)
#pragma unroll
    for (int ni = 0; ni < 4; ++ni) acc[mi][ni] = z8;

  const float* ap[4];
  const float* bp[4];
#pragma unroll
  for (int mi = 0; mi < 4; ++mi)
    ap[mi] = A + (size_t)(Moff + mi*16 + ln) * K + half*2;
#pragma unroll
  for (int ni = 0; ni < 4; ++ni)
    bp[ni] = W + (size_t)(Noff + ni*16 + ln) * K + half*2;

  for (int k = 0; k < K; k += 4) {
    v2f av[4], bv[4];
#pragma unroll
    for (int mi = 0; mi < 4; ++mi) { av[mi] = ldg_v2f(ap[mi]); ap[mi] += 4; }
#pragma unroll
    for (int ni = 0; ni < 4; ++ni) { bv[ni] = ldg_v2f(bp[ni]); bp[ni] += 4; }
#pragma unroll
    for (int mi = 0; mi < 4; ++mi)
#pragma unroll
      for (int ni = 0; ni < 4; ++ni)
        acc[mi][ni] = wmma4(av[mi], bv[ni], acc[mi][ni]);
  }

#pragma unroll
  for (int ni = 0; ni < 4; ++ni) {
    const int n = Noff + ni*16 + ln;
    const float bvs = ldg_f(bias + n);
#pragma unroll
    for (int mi = 0; mi < 4; ++mi) {
      const int row0 = Moff + mi*16 + half*8;
#pragma unroll
      for (int i = 0; i < 8; ++i) {
        float v = acc[mi][ni][i] + bvs;
        if (act) v = tanhf(v);
        stg_f(C + (size_t)(row0 + i) * N + n, v);
      }
    }
  }
}

// ---------------------------------------------------------------------------
// GRU recurrence, one workgroup (1024 thr = 32 waves) per direction.
// h [64 x 384] lives in LDS (dynamic, 64*388*4 = 99328 B).
// Per step: gh = h @ W_hh^T  (M=64,N=1152,K=384) -> 96 output tiles of which
// each wave owns 3 "tasks"; one task = (m-tile, col-tile j) covering the r,z,n
// gate columns so the A operand (from LDS) is shared by 3 WMMAs.
// gx rows are laid out as [b*T + t][1152]. Output y rows [b*T + t][768],
// forward dir writes cols 0..383, backward 384..767.
// ---------------------------------------------------------------------------
__global__ __launch_bounds__(1024) void gru_scan_kernel(
    const float* __restrict__ gx_f, const float* __restrict__ gx_b,
    const float* __restrict__ whh_f, const float* __restrict__ whh_b,
    const float* __restrict__ bhh_f, const float* __restrict__ bhh_b,
    float* __restrict__ y)
{
  extern __shared__ float hsh[];             // [64][HPAD]
  const int dir = blockIdx.x;                // 0 = forward, 1 = backward
  const float* gx  = dir ? gx_b  : gx_f;
  const float* whh = dir ? whh_b : whh_f;
  const float* bhh = dir ? bhh_b : bhh_f;
  const int col_off = dir ? HID : 0;

  const int tid  = threadIdx.x;
  const int wave = tid >> 5;
  const int lane = tid & 31;
  const int ln   = lane & 15;
  const int half = lane >> 4;

  for (int i = tid; i < BATCH * HPAD; i += 1024) hsh[i] = 0.0f;
  __syncthreads();

  const v8f z8 = {0,0,0,0,0,0,0,0};

  for (int s = 0; s < T_SEQ; ++s) {
    const int t = dir ? (T_SEQ - 1 - s) : s;
    v8f hnew[3];

#pragma unroll
    for (int tk = 0; tk < 3; ++tk) {
      const int task = wave * 3 + tk;        // 0..95
      const int j  = task % 24;              // hidden col-tile (16 cols)
      const int mt = task / 24;              // batch m-tile (16 rows)

      v8f ar = z8, az = z8, an = z8;
      const float* wr = whh + (size_t)(          j*16 + ln) * HID + half*2;
      const float* wz = whh + (size_t)(HID     + j*16 + ln) * HID + half*2;
      const float* wn = whh + (size_t)(2*HID   + j*16 + ln) * HID + half*2;
      const float* ha = hsh + (mt*16 + ln) * HPAD + half*2;

      for (int k = 0; k < HID; k += 4) {
        v2f a  = *(const v2f*)ha; ha += 4;   // LDS read (ds_load_b64)
        v2f br = ldg_v2f(wr); wr += 4;
        v2f bz = ldg_v2f(wz); wz += 4;
        v2f bn = ldg_v2f(wn); wn += 4;
        ar = wmma4(a, br, ar);
        az = wmma4(a, bz, az);
        an = wmma4(a, bn, an);
      }

      const int col   = j*16 + ln;           // hidden column 0..383
      const int brow0 = mt*16 + half*8;      // batch row base for acc idx 0
      const float bhr = ldg_f(bhh + col);
      const float bhz = ldg_f(bhh + HID   + col);
      const float bhn = ldg_f(bhh + 2*HID + col);

      v8f hn;
#pragma unroll
      for (int i = 0; i < 8; ++i) {
        const int b = brow0 + i;
        const size_t base = ((size_t)b * T_SEQ + t) * G3;
        const float gr = ldg_f(gx + base + col);
        const float gz = ldg_f(gx + base + HID   + col);
        const float gn = ldg_f(gx + base + 2*HID + col);
        const float r  = 1.0f / (1.0f + expf(-(gr + ar[i] + bhr)));
        const float zt = 1.0f / (1.0f + expf(-(gz + az[i] + bhz)));
        const float nn = tanhf(gn + r * (an[i] + bhn));
        const float hold = hsh[b * HPAD + col];
        hn[i] = (1.0f - zt) * nn + zt * hold;
      }
      hnew[tk] = hn;
    }

    __syncthreads();                         // everyone done reading h
#pragma unroll
    for (int tk = 0; tk < 3; ++tk) {
      const int task = wave * 3 + tk;
      const int j  = task % 24;
      const int mt = task / 24;
      const int col   = j*16 + ln;
      const int brow0 = mt*16 + half*8;
#pragma unroll
      for (int i = 0; i < 8; ++i) {
        const int b = brow0 + i;
        hsh[b * HPAD + col] = hnew[tk][i];
        stg_f(y + ((size_t)b * T_SEQ + t) * DIM + col_off + col, hnew[tk][i]);
      }
    }
    __syncthreads();                         // h updated for next step
  }
}

// ---------------------------------------------------------------------------
// Attention pooling: one block per batch element, 256 threads (one per t).
// logits -> softmax over T -> weighted sum of hidden -> out[b][768].
// upw_w is staged in LDS once per block.
// ---------------------------------------------------------------------------
__global__ __launch_bounds__(256) void attn_pool_kernel(
    const float* __restrict__ u, const float* __restrict__ hid,
    const float* __restrict__ upw_w, const float* __restrict__ upw_b,
    float* __restrict__ out)
{
  __shared__ float red[256];
  __shared__ float wsh[DIM];
  const int b   = blockIdx.x;
  const int tid = threadIdx.x;

  for (int d = tid; d < DIM; d += 256) wsh[d] = ldg_f(upw_w + d);
  __syncthreads();

  const float* up = u + ((size_t)b * T_SEQ + tid) * DIM;
  float lg = ldg_f(upw_b);
  for (int d = 0; d < DIM; d += 2) {
    v2f uv = ldg_v2f(up + d);
    lg += uv.x * wsh[d] + uv.y * wsh[d + 1];
  }

  red[tid] = lg; __syncthreads();
  for (int s = 128; s > 0; s >>= 1) {
    if (tid < s) red[tid] = fmaxf(red[tid], red[tid + s]);
    __syncthreads();
  }
  const float mx = red[0]; __syncthreads();
  const float e = expf(lg - mx);
  red[tid] = e; __syncthreads();
  for (int s = 128; s > 0; s >>= 1) {
    if (tid < s) red[tid] += red[tid + s];
    __syncthreads();
  }
  const float inv = 1.0f / red[0]; __syncthreads();
  red[tid] = e * inv;                        // alpha[t]
  __syncthreads();

  for (int d = tid; d < DIM; d += 256) {
    float accv = 0.0f;
    const float* hp = hid + (size_t)b * T_SEQ * DIM + d;
    for (int t2 = 0; t2 < T_SEQ; ++t2) accv += red[t2] * ldg_f(hp + (size_t)t2 * DIM);
    out[(size_t)b * DIM + d] = accv;
  }
}

// ---------------------------------------------------------------------------
// Workspace layout (floats):
//   gxA : 16384*1152  (layer-L forward input gates)
//   gxB : 16384*1152  (layer-L backward input gates)
//   xb  : 16384*768   (layer-0 output, later reused as u)
//   hb  : 16384*768   (layer-1 output = hidden)
// Total = 251,658,240 bytes.
// Activation rows everywhere are indexed as [b*T + t] so that Sentences
// ([B,T,I] contiguous) is directly a [16384 x 768] GEMM operand.
// ---------------------------------------------------------------------------
extern "C" void kernel_launch(void* const* d_in, const int* in_sizes, int n_in,
                              void* d_out, int out_size, void* d_ws, size_t ws_size,
                              hipStream_t stream)
{
  (void)in_sizes; (void)n_in; (void)out_size; (void)ws_size;

  const float* S        = (const float*)d_in[0];
  const float* w_ih_l0f = (const float*)d_in[1];
  const float* w_hh_l0f = (const float*)d_in[2];
  const float* b_ih_l0f = (const float*)d_in[3];
  const float* b_hh_l0f = (const float*)d_in[4];
  const float* w_ih_l0b = (const float*)d_in[5];
  const float* w_hh_l0b = (const float*)d_in[6];
  const float* b_ih_l0b = (const float*)d_in[7];
  const float* b_hh_l0b = (const float*)d_in[8];
  const float* w_ih_l1f = (const float*)d_in[9];
  const float* w_hh_l1f = (const float*)d_in[10];
  const float* b_ih_l1f = (const float*)d_in[11];
  const float* b_hh_l1f = (const float*)d_in[12];
  const float* w_ih_l1b = (const float*)d_in[13];
  const float* w_hh_l1b = (const float*)d_in[14];
  const float* b_ih_l1b = (const float*)d_in[15];
  const float* b_hh_l1b = (const float*)d_in[16];
  const float* fc_w     = (const float*)d_in[17];
  const float* fc_b     = (const float*)d_in[18];
  const float* upw_w    = (const float*)d_in[19];
  const float* upw_b    = (const float*)d_in[20];

  float* ws = (float*)d_ws;
  const size_t GX = (size_t)BATCH * T_SEQ * G3;
  const size_t XT = (size_t)BATCH * T_SEQ * DIM;
  float* gxA = ws;
  float* gxB = gxA + GX;
  float* xb  = gxB + GX;
  float* hb  = xb + XT;

  const dim3 blk(256);
  const dim3 g_gates(64, 9);   // M=16384/256, N=1152/128
  const dim3 g_fc(64, 6);      // M=16384/256, N= 768/128
  const size_t scan_lds = (size_t)BATCH * HPAD * sizeof(float);

  // Layer 0 input gates (fwd + bwd)
  gemm_bias_act<<<g_gates, blk, 0, stream>>>(S, w_ih_l0f, b_ih_l0f, gxA, DIM, G3, 0);
  gemm_bias_act<<<g_gates, blk, 0, stream>>>(S, w_ih_l0b, b_ih_l0b, gxB, DIM, G3, 0);
  // Layer 0 recurrence -> xb [b*T+t][768]
  gru_scan_kernel<<<dim3(2), dim3(1024), scan_lds, stream>>>(
      gxA, gxB, w_hh_l0f, w_hh_l0b, b_hh_l0f, b_hh_l0b, xb);
  // Layer 1 input gates
  gemm_bias_act<<<g_gates, blk, 0, stream>>>(xb, w_ih_l1f, b_ih_l1f, gxA, DIM, G3, 0);
  gemm_bias_act<<<g_gates, blk, 0, stream>>>(xb, w_ih_l1b, b_ih_l1b, gxB, DIM, G3, 0);
  // Layer 1 recurrence -> hb (hidden)
  gru_scan_kernel<<<dim3(2), dim3(1024), scan_lds, stream>>>(
      gxA, gxB, w_hh_l1f, w_hh_l1b, b_hh_l1f, b_hh_l1b, hb);
  // u = tanh(hidden @ fc_w^T + fc_b) -> reuse xb
  gemm_bias_act<<<g_fc, blk, 0, stream>>>(hb, fc_w, fc_b, xb, DIM, DIM, 1);
  // attention pooling -> out [64 x 768]
  attn_pool_kernel<<<dim3(64), dim3(256), 0, stream>>>(xb, hb, upw_w, upw_b, (float*)d_out);
}